// StepAttractor_76957224010415
// MI455X (gfx1250) — compile-verified
//
#include <hip/hip_runtime.h>

#define DIM 4096

typedef float v2f __attribute__((ext_vector_type(2)));
typedef float v4f __attribute__((ext_vector_type(4)));
typedef float v8f __attribute__((ext_vector_type(8)));

// One-time init: x -> zero-padded transposed xTpad[DIM][16]; zero hsT, hs_ws, xM.
__global__ __launch_bounds__(256) void init_xT(const float* __restrict__ x,
                                               float* __restrict__ xTpad,
                                               float* __restrict__ hsT,
                                               float* __restrict__ hs_ws,
                                               float* __restrict__ xM) {
    int i = blockIdx.x * blockDim.x + threadIdx.x;
    if (i < DIM) {
        for (int b = 0; b < 16; ++b)
            xTpad[i * 16 + b] = (b < 4) ? x[b * DIM + i] : 0.0f;
        for (int b = 0; b < 4; ++b) {
            hsT[i * 4 + b]      = 0.0f;
            hs_ws[b * DIM + i]  = 0.0f;
            xM[b * DIM + i]     = 0.0f;
        }
    }
}

__device__ __forceinline__ float dot4(v4f a, v4f b) {
    return a.x * b.x + a.y * b.y + a.z * b.z + a.w * b.w;
}

// Fused pass over M: Mdst = rr*Msrc + lr*(hs^T hs), and xM += x @ Mdst.
// Grid (DIM/32, 4): blockIdx.x = 32-column group (two 16-col WMMA tiles:
// even cols jbase+2n, odd cols jbase+2n+1 -> b64 M access per lane),
// blockIdx.y = row quarter. 4 waves per WG split the quarter's rows.
__global__ __launch_bounds__(128) void fused_step(
    const float* __restrict__ Msrc, float* __restrict__ Mdst,
    const float* __restrict__ xTpad, // [DIM][16], cols 4..15 zero
    const float* __restrict__ hs,    // [4][DIM]  column factors
    const float* __restrict__ hsT,   // [DIM][4]  row factors
    float* __restrict__ xM,          // [4][DIM]  accumulated via atomics
    float rr, float lr)
{
    const int lane  = threadIdx.x & 31;
    const int wave  = threadIdx.x >> 5;     // 0..3
    const int col   = lane & 15;
    const int half  = lane >> 4;            // 0 or 1
    const int jbase = blockIdx.x * 32;
    const int je    = jbase + 2 * col;      // even-tile column
    const int jo    = je + 1;               // odd-tile column

    // Column factors, pre-scaled by lr (lr=0 on first pass -> exact copy of M).
    v4f ve, vo;
    ve.x = hs[0 * DIM + je] * lr;  vo.x = hs[0 * DIM + jo] * lr;
    ve.y = hs[1 * DIM + je] * lr;  vo.y = hs[1 * DIM + jo] * lr;
    ve.z = hs[2 * DIM + je] * lr;  vo.z = hs[2 * DIM + jo] * lr;
    ve.w = hs[3 * DIM + je] * lr;  vo.w = hs[3 * DIM + jo] * lr;

    v8f ce = {};   // accumulator, even columns
    v8f co = {};   // accumulator, odd columns

    const int rowsPerWave = DIM / 16;                       // 256
    const int i0 = (blockIdx.y * 4 + wave) * rowsPerWave;
    const int i1 = i0 + rowsPerWave;

    for (int ib = i0; ib < i1; ib += 4) {
        const int r0 = ib + half * 2;       // A/B layout: vgpr0 = K0|K2
        const int r1 = r0 + 1;              //             vgpr1 = K1|K3

        v2f m0 = *(const v2f*)(Msrc + (size_t)r0 * DIM + je);
        v2f m1 = *(const v2f*)(Msrc + (size_t)r1 * DIM + je);

        v4f u0 = *(const v4f*)(hsT + (size_t)r0 * 4);
        v4f u1 = *(const v4f*)(hsT + (size_t)r1 * 4);

        m0.x = rr * m0.x + dot4(u0, ve);
        m0.y = rr * m0.y + dot4(u0, vo);
        m1.x = rr * m1.x + dot4(u1, ve);
        m1.y = rr * m1.y + dot4(u1, vo);

        *(v2f*)(Mdst + (size_t)r0 * DIM + je) = m0;
        *(v2f*)(Mdst + (size_t)r1 * DIM + je) = m1;

        // A operand (shared by both WMMAs): M-index = lane&15, K = half*2(+1).
        v2f a;
        a.x = xTpad[(size_t)r0 * 16 + col];
        a.y = xTpad[(size_t)r1 * 16 + col];

        v2f be; be.x = m0.x; be.y = m1.x;   // even-tile B (4x16)
        v2f bo; bo.x = m0.y; bo.y = m1.y;   // odd-tile  B (4x16)

        ce = __builtin_amdgcn_wmma_f32_16x16x4_f32(
                 false, a, false, be, (short)0, ce, false, false);
        co = __builtin_amdgcn_wmma_f32_16x16x4_f32(
                 false, a, false, bo, (short)0, co, false, false);
    }

    // C rows 0..3 (lanes 0..15, vgprs 0..3) hold partial xM for this wave.
    __shared__ float red[4][4][16][2];
    if (half == 0) {
        for (int b = 0; b < 4; ++b) {
            red[wave][b][col][0] = ce[b];
            red[wave][b][col][1] = co[b];
        }
    }
    __syncthreads();
    if (wave == 0 && half == 0) {
        for (int b = 0; b < 4; ++b) {
            float se = red[0][b][col][0] + red[1][b][col][0] +
                       red[2][b][col][0] + red[3][b][col][0];
            float so = red[0][b][col][1] + red[1][b][col][1] +
                       red[2][b][col][1] + red[3][b][col][1];
            atomicAdd(&xM[b * DIM + je], se);
            atomicAdd(&xM[b * DIM + jo], so);
        }
    }
}

// hs = normalize(leaky_relu(DECAY*hs + xM)); writes row-major + transposed copies,
// and zeroes xM in place for the next accumulation pass.
__global__ __launch_bounds__(512) void hs_update(
    const float* __restrict__ hs_in,   // [4][DIM]
    float* __restrict__ xM,            // [4][DIM], zeroed after use
    float* __restrict__ hs_out,        // [4][DIM] (may alias hs_in)
    float* __restrict__ hsT_out)       // [DIM][4]
{
    const int tid = threadIdx.x;
    __shared__ float part[4][512];

    float acc[4] = {0.f, 0.f, 0.f, 0.f};
    for (int b = 0; b < 4; ++b) {
        for (int jj = tid; jj < DIM; jj += 512) {
            float pre = 0.9f * hs_in[b * DIM + jj] + xM[b * DIM + jj];
            float act = (pre >= 0.0f) ? pre : 0.01f * pre;
            acc[b] += act * act;
        }
    }
    for (int b = 0; b < 4; ++b) part[b][tid] = acc[b];
    __syncthreads();
    for (int s = 256; s > 0; s >>= 1) {
        if (tid < s)
            for (int b = 0; b < 4; ++b) part[b][tid] += part[b][tid + s];
        __syncthreads();
    }

    for (int b = 0; b < 4; ++b) {
        float mx = fmaxf(sqrtf(part[b][0]), 1e-12f);
        for (int jj = tid; jj < DIM; jj += 512) {
            float xv  = xM[b * DIM + jj];
            float pre = 0.9f * hs_in[b * DIM + jj] + xv;
            float act = (pre >= 0.0f) ? pre : 0.01f * pre;
            float h = act / mx;
            hs_out[b * DIM + jj] = h;
            hsT_out[jj * 4 + b]  = h;
            xM[b * DIM + jj] = 0.0f;   // ready for next pass's atomics
        }
    }
}

extern "C" void kernel_launch(void* const* d_in, const int* in_sizes, int n_in,
                              void* d_out, int out_size, void* d_ws, size_t ws_size,
                              hipStream_t stream) {
    const float* x   = (const float*)d_in[0];
    const float* M0  = (const float*)d_in[1];
    const float* hs0 = (const float*)d_in[2];
    float* out = (float*)d_out;

    char* ws = (char*)d_ws;
    const size_t MBYTES = (size_t)DIM * DIM * sizeof(float);   // 64 MiB
    float* M_ws  = (float*)ws;
    float* xM    = (float*)(ws + MBYTES);
    float* hs_ws = (float*)(ws + MBYTES + (64 << 10));
    float* hsT   = (float*)(ws + MBYTES + (128 << 10));
    float* xTpad = (float*)(ws + MBYTES + (192 << 10));        // 256 KiB

    init_xT<<<DIM / 256, 256, 0, stream>>>(x, xTpad, hsT, hs_ws, xM);

    // Pass 0: M_ws = M0 (rr=1, lr=0), xM = x @ M0.
    fused_step<<<dim3(DIM / 32, 4), 128, 0, stream>>>(M0, M_ws, xTpad, hs_ws, hsT,
                                                      xM, 1.0f, 0.0f);

    for (int t = 1; t <= 16; ++t) {
        const float* hin  = (t == 1)  ? hs0 : hs_ws;
        float*       hout = (t == 16) ? out : hs_ws;
        hs_update<<<1, 512, 0, stream>>>(hin, xM, hout, hsT);
        if (t < 16) {
            // M_ws = RR*M_ws + LR*hs^T hs, and xM = x @ M_new for the next step.
            fused_step<<<dim3(DIM / 32, 4), 128, 0, stream>>>(M_ws, M_ws, xTpad,
                                                              hs_ws, hsT, xM,
                                                              0.99f, 0.01f);
        }
    }
}